// equi_conv3_21603685499532
// MI455X (gfx1250) — compile-verified
//
#include <hip/hip_runtime.h>
#include <math.h>
#include <stdint.h>

typedef __attribute__((ext_vector_type(2))) float v2f;
typedef __attribute__((ext_vector_type(8))) float v8f;

namespace {
constexpr int kBS   = 2;
constexpr int kP    = 1024;
constexpr int kN    = 16;
constexpr int kDin  = 64;
constexpr int kDout = 32;
constexpr int kK    = 7;
constexpr int kR    = 60;
constexpr int kDK   = kDout * kK;          // 224
constexpr int kMRows = kDK * 3 + kDout;    // 704 = 672 U rows + 32 center rows
constexpr int kNCols = kBS * kP * 3;       // 6144
constexpr int kUPad = 12;                  // U row padded 9 -> 12 floats (48B)

// workspace layout (float offsets); total ~6.37M floats (~25.5 MB)
constexpr size_t OFF_WP   = 0;                                           // 704*64
constexpr size_t OFF_KDN  = OFF_WP  + (size_t)kMRows * kDin;             // 224*3
constexpr size_t OFF_NDN  = OFF_KDN + (size_t)kDK * 3;                   // 2*1024*16*3
constexpr size_t OFF_UPK  = OFF_NDN + (size_t)kBS * kP * kN * 3;         // 2*1024*224*12
constexpr size_t OFF_CTR  = OFF_UPK + (size_t)kBS * kP * kDK * kUPad;    // 32*6144
constexpr size_t OFF_FO2  = OFF_CTR + (size_t)kDout * kNCols;            // 2*32*7*1024
constexpr size_t OFF_RIDX = OFF_FO2 + (size_t)kBS * kDout * kK * kP;     // 65536 ints
constexpr float kNegInf = -3.402823466e38f;
}

// ---------------------------------------------------------------------------
// Kernel 1: repack W' = [W(d,k,j) rows ; W_center rows] as 704x64, normalize
// kernel directions, compute normalized neighbor directions ndn.
// ---------------------------------------------------------------------------
__global__ void prep_kernel(const int* __restrict__ nbrI,
                            const float* __restrict__ verts,
                            const float* __restrict__ W,
                            const float* __restrict__ Wc,
                            const float* __restrict__ dirs,
                            float* __restrict__ ws) {
  int tid = blockIdx.x * blockDim.x + threadIdx.x;
  float* Wp  = ws + OFF_WP;
  float* kdn = ws + OFF_KDN;
  float* ndn = ws + OFF_NDN;
  if (tid < kMRows * kDin) {                       // 45056
    int row = tid / kDin, c = tid % kDin;
    float v;
    if (row < kDK * 3) {
      int d = row / 21, rem = row % 21, k = rem / 3, j = rem % 3;
      v = W[(((size_t)d * kDin + c) * kK + k) * 3 + j];
    } else {
      v = Wc[(size_t)(row - kDK * 3) * kDin + c];
    }
    Wp[(size_t)row * kDin + c] = v;
    return;
  }
  tid -= kMRows * kDin;
  if (tid < kDK) {                                 // 224: normalize directions
    float x = dirs[tid*3], y = dirs[tid*3+1], z = dirs[tid*3+2];
    float s = fmaxf(sqrtf(x*x + y*y + z*z), 1e-12f);
    kdn[tid*3] = x/s; kdn[tid*3+1] = y/s; kdn[tid*3+2] = z/s;
    return;
  }
  tid -= kDK;
  if (tid < kBS * kP * kN) {                       // 32768: ndn
    int b = tid / (kP * kN);
    int p = (tid / kN) % kP;
    int nb = nbrI[tid];
    const float* vc = verts + ((size_t)b * kP + p) * 3;
    const float* vn = verts + ((size_t)b * kP + nb) * 3;
    float x = vn[0]-vc[0], y = vn[1]-vc[1], z = vn[2]-vc[2];
    float s = fmaxf(sqrtf(x*x + y*y + z*z), 1e-12f);
    ndn[tid*3] = x/s; ndn[tid*3+1] = y/s; ndn[tid*3+2] = z/s;
  }
}

// ---------------------------------------------------------------------------
// Kernel 2: WMMA f32 GEMM  Utot[704 x 6144] = Wp[704 x 64] @ fm'[64 x 6144]
// fm'[c][(b,p,i)] read directly from feature_map. Output scattered into
// Upk[b][p][k][d][0..8] (rows padded to 12 floats, 48B aligned) and
// Ctr[d][(b,p,i)]. One 16x16 tile/wave; K-loop = 16x V_WMMA_F32_16X16X4_F32.
// ---------------------------------------------------------------------------
__global__ __launch_bounds__(256) void gemm_wmma_kernel(
    const float* __restrict__ fm, float* __restrict__ ws) {
  const float* Wp = ws + OFF_WP;
  float* Upk = ws + OFF_UPK;
  float* Ctr = ws + OFF_CTR;

  int wave = threadIdx.x >> 5;
  int lane = threadIdx.x & 31;
  int gid  = blockIdx.x * 8 + wave;         // tile id: exactly 44*384 = 16896
  int mt = gid % 44;
  int nt = gid / 44;

  int half = lane >> 4;                     // K split per documented layout
  int l16  = lane & 15;
  int m = mt * 16 + l16;
  int n = nt * 16 + l16;
  int bsel = (n >= kP * 3) ? 1 : 0;
  // fm flat index: b*196608 + c*3072 + p*3 + i ;  n = b*3072 + p*3 + i
  int fmbase = n + bsel * (kDin - 1) * (kP * 3);

  v8f acc = {0.f,0.f,0.f,0.f,0.f,0.f,0.f,0.f};
  const float* arow = Wp + (size_t)m * kDin;
#pragma unroll
  for (int kk = 0; kk < kDin; kk += 4) {
    v2f a, b2;
    a.x  = arow[kk + half*2];               // A: VGPR0=K0/K2, VGPR1=K1/K3
    a.y  = arow[kk + half*2 + 1];
    b2.x = fm[fmbase + (kk + half*2)     * (kP*3)];
    b2.y = fm[fmbase + (kk + half*2 + 1) * (kP*3)];
    acc = __builtin_amdgcn_wmma_f32_16x16x4_f32(false, a, false, b2,
                                                (short)0, acc, false, false);
  }
  int pcol = n - bsel * (kP*3);
  int p = pcol / 3, ii = pcol % 3;
#pragma unroll
  for (int v = 0; v < 8; ++v) {             // D: VGPR v = rows v / v+8
    int mo = mt*16 + v + half*8;
    float val = acc[v];
    if (mo < kDK*3) {
      int d = mo / 21, rem = mo % 21, k = rem / 3, j = rem % 3;
      Upk[((size_t)(bsel*kP + p) * kDK + k*kDout + d) * kUPad + j*3 + ii] = val;
    } else {
      Ctr[(size_t)(mo - kDK*3) * kNCols + n] = val;
    }
  }
}

// ---------------------------------------------------------------------------
// Kernel 3: rotation search. Block = (b,p). Stage neighbor U slices per-k into
// LDS with GLOBAL_LOAD_ASYNC_TO_LDS_B128 (ASYNCcnt / s_wait_asynccnt), then
// threads own (d,r) pairs; s[d,r] = sum_k max_n (fo * theta); argmax_r.
// Bin-structured argmax of the reference == global first-argmax over r.
// ---------------------------------------------------------------------------
__global__ __launch_bounds__(256) void search_kernel(
    const int* __restrict__ nbrI, const float* __restrict__ RsG,
    float* __restrict__ ws) {
  __shared__ float sR[kR*9];            // 540
  __shared__ float sKdn[kDK*3];         // 672
  __shared__ float sNdn[kN*3];          // 48
  __shared__ int   sNbr[kN];
  __shared__ float sUk[kN*kDout*kUPad] __attribute__((aligned(16))); // 6144 f (24KB)
  __shared__ float sS[kDout*kR];        // 1920

  const float* kdn = ws + OFF_KDN;
  const float* ndn = ws + OFF_NDN;
  const float* Upk = ws + OFF_UPK;
  int* ridx = (int*)(ws + OFF_RIDX);

  int bp  = blockIdx.x;                 // b*1024 + p
  int tid = threadIdx.x;
  int b   = bp >> 10;

  for (int i = tid; i < kR*9; i += 256)  sR[i] = RsG[i];
  for (int i = tid; i < kDK*3; i += 256) sKdn[i] = kdn[i];
  for (int i = tid; i < kN*3; i += 256)  sNdn[i] = ndn[(size_t)bp*kN*3 + i];
  if (tid < kN) sNbr[tid] = nbrI[bp*kN + tid];
  __syncthreads();

  unsigned sUkBase = (unsigned)(uintptr_t)(&sUk[0]);
  constexpr int kChunks = kN * (kDout * kUPad * 4 / 16);  // 16 * 96 = 1536

  float accq[8];
#pragma unroll
  for (int q = 0; q < 8; ++q) accq[q] = 0.f;

  for (int k = 0; k < kK; ++k) {
    __syncthreads();
    // async-copy this k's neighbor U slices: per neighbor 32*12 floats = 1536B
    // contiguous and 16B aligned -> 96 b128 chunks; 1536 chunks total = 6 full
    // 256-lane iterations.
    for (int c = tid; c < kChunks; c += 256) {
      int nn = c / 96, q16 = c % 96;
      unsigned goff = (unsigned)(((b*kP + sNbr[nn]) * (kDK*kUPad) +
                                  k * (kDout*kUPad)) * 4) + (unsigned)(q16 * 16);
      unsigned loff = sUkBase + (unsigned)(nn * (kDout*kUPad*4)) + (unsigned)(q16 * 16);
      asm volatile("global_load_async_to_lds_b128 %0, %1, %2"
                   :: "v"(loff), "v"(goff), "s"(Upk)
                   : "memory");
    }
    asm volatile("s_wait_asynccnt 0x0" ::: "memory");
    __syncthreads();
#pragma unroll
    for (int q = 0; q < 8; ++q) {
      int pair = tid + 256*q;
      if (pair >= kDout*kR) break;
      int d = pair / kR, r = pair % kR;
      const float* Rg = &sR[r*9];
      const float* kv = &sKdn[(d*kK + k)*3];
      float kd0 = kv[0], kd1 = kv[1], kd2 = kv[2];
      // kd_ro[i] = sum_j kd[j]*R[i][j]
      float o0 = kd0*Rg[0] + kd1*Rg[1] + kd2*Rg[2];
      float o1 = kd0*Rg[3] + kd1*Rg[4] + kd2*Rg[5];
      float o2 = kd0*Rg[6] + kd1*Rg[7] + kd2*Rg[8];
      float mx = kNegInf;
      for (int nn = 0; nn < kN; ++nn) {
        const float* Up = &sUk[(nn*kDout + d)*kUPad];   // U[j*3+i], 48B-aligned
        // fo = sum_{j,i} U[j,i] * R[i,j]
        float fo = Up[0]*Rg[0] + Up[1]*Rg[3] + Up[2]*Rg[6]
                 + Up[3]*Rg[1] + Up[4]*Rg[4] + Up[5]*Rg[7]
                 + Up[6]*Rg[2] + Up[7]*Rg[5] + Up[8]*Rg[8];
        float th = o0*sNdn[nn*3] + o1*sNdn[nn*3+1] + o2*sNdn[nn*3+2];
        mx = fmaxf(mx, fo*th);
      }
      accq[q] += mx;
    }
  }
  __syncthreads();
#pragma unroll
  for (int q = 0; q < 8; ++q) {
    int pair = tid + 256*q;
    if (pair < kDout*kR) sS[pair] = accq[q];
  }
  __syncthreads();
  if (tid < kDout) {
    float best = kNegInf; int bi = 0;
    for (int r = 0; r < kR; ++r) {          // strict > keeps first max
      float v = sS[tid*kR + r];
      if (v > best) { best = v; bi = r; }
    }
    ridx[((size_t)b*kDout + tid)*kP + (bp & 1023)] = bi;
  }
}

// ---------------------------------------------------------------------------
// Kernel 4: fo2[b,d,k,p] = <U[b,p,d,k], Rs[r_index[b,d,p]]>
// ---------------------------------------------------------------------------
__global__ void fo2_kernel(const float* __restrict__ RsG,
                           float* __restrict__ ws) {
  int idx = blockIdx.x * blockDim.x + threadIdx.x;   // ((b*32+d)*7+k)*1024+p
  const float* Upk = ws + OFF_UPK;
  float* fo2 = ws + OFF_FO2;
  const int* ridx = (const int*)(ws + OFF_RIDX);
  int p = idx & 1023;
  int k = (idx >> 10) % kK;
  int d = (idx / (kK << 10)) % kDout;
  int b = idx / (kDout * kK * kP);
  int r = ridx[((size_t)b*kDout + d)*kP + p];
  const float* Rg = RsG + r*9;
  const float* Up = Upk + ((size_t)(b*kP + p)*kDK + k*kDout + d)*kUPad;
  fo2[idx] = Up[0]*Rg[0] + Up[1]*Rg[3] + Up[2]*Rg[6]
           + Up[3]*Rg[1] + Up[4]*Rg[4] + Up[5]*Rg[7]
           + Up[6]*Rg[2] + Up[7]*Rg[5] + Up[8]*Rg[8];
}

// ---------------------------------------------------------------------------
// Kernel 5: theta2 = relu(kd_ro2 . ndn); kaw2 = max_n gathered2*theta2;
// out = sum_k kaw2 * kd_ro2 + center.
// ---------------------------------------------------------------------------
__global__ void final_kernel(const int* __restrict__ nbrI,
                             const float* __restrict__ RsG,
                             const float* __restrict__ ws,
                             float* __restrict__ out) {
  int idx = blockIdx.x * blockDim.x + threadIdx.x;   // (b*32+d)*1024+p
  const float* kdn = ws + OFF_KDN;
  const float* ndn = ws + OFF_NDN;
  const float* Ctr = ws + OFF_CTR;
  const float* fo2 = ws + OFF_FO2;
  const int* ridx = (const int*)(ws + OFF_RIDX);
  int p = idx & 1023;
  int d = (idx >> 10) & 31;
  int b = idx >> 15;
  int r = ridx[idx];
  float Rg[9];
#pragma unroll
  for (int i = 0; i < 9; ++i) Rg[i] = RsG[r*9 + i];
  float o0 = 0.f, o1 = 0.f, o2 = 0.f;
  int bp = b*kP + p;
  for (int k = 0; k < kK; ++k) {
    const float* kv = &kdn[(d*kK + k)*3];
    float c0 = kv[0]*Rg[0] + kv[1]*Rg[1] + kv[2]*Rg[2];
    float c1 = kv[0]*Rg[3] + kv[1]*Rg[4] + kv[2]*Rg[5];
    float c2 = kv[0]*Rg[6] + kv[1]*Rg[7] + kv[2]*Rg[8];
    float kaw = kNegInf;
    for (int nn = 0; nn < kN; ++nn) {
      int nb = nbrI[bp*kN + nn];
      const float* nd = &ndn[((size_t)bp*kN + nn)*3];
      float th = fmaxf(c0*nd[0] + c1*nd[1] + c2*nd[2], 0.f);
      float g = fo2[((size_t)(b*kDout + d)*kK + k)*kP + nb];
      kaw = fmaxf(kaw, g*th);
    }
    o0 += kaw*c0; o1 += kaw*c1; o2 += kaw*c2;
  }
  int col = b*(kP*3) + p*3;
  o0 += Ctr[(size_t)d*kNCols + col];
  o1 += Ctr[(size_t)d*kNCols + col + 1];
  o2 += Ctr[(size_t)d*kNCols + col + 2];
  out[(size_t)idx*3]     = o0;
  out[(size_t)idx*3 + 1] = o1;
  out[(size_t)idx*3 + 2] = o2;
}

extern "C" void kernel_launch(void* const* d_in, const int* in_sizes, int n_in,
                              void* d_out, int out_size, void* d_ws, size_t ws_size,
                              hipStream_t stream) {
  const int*   nbrI  = (const int*)d_in[0];
  const float* verts = (const float*)d_in[1];
  const float* fm    = (const float*)d_in[2];
  const float* W     = (const float*)d_in[3];
  const float* Wc    = (const float*)d_in[4];
  const float* dirs  = (const float*)d_in[5];
  const float* RsG   = (const float*)d_in[6];
  float* ws  = (float*)d_ws;   // needs ~25.5 MB
  float* out = (float*)d_out;

  int prepThreads = kMRows*kDin + kDK + kBS*kP*kN;   // 78048
  prep_kernel<<<(prepThreads + 255)/256, 256, 0, stream>>>(nbrI, verts, W, Wc, dirs, ws);
  gemm_wmma_kernel<<<(44*384)/8, 256, 0, stream>>>(fm, ws);
  search_kernel<<<kBS*kP, 256, 0, stream>>>(nbrI, RsG, ws);
  fo2_kernel<<<(kBS*kDout*kK*kP)/256, 256, 0, stream>>>(RsG, ws);
  final_kernel<<<(kBS*kDout*kP)/256, 256, 0, stream>>>(nbrI, RsG, ws, out);
}